// LNN_50302656971188
// MI455X (gfx1250) — compile-verified
//
#include <hip/hip_runtime.h>

typedef float v2f __attribute__((ext_vector_type(2)));
typedef float v4f __attribute__((ext_vector_type(4)));
typedef float v8f __attribute__((ext_vector_type(8)));

#define SJ 132   // padded row stride (floats) for per-wave sample buffers
#define SP 130   // padded row stride (float2) for paired B matrices

// ---- LDS layout (float offsets) ----
#define OFF_W2TP 0                         // W2Tp[k/2][j] = {W2[j][k],W2[j][k+1]}
#define OFF_W2NP (64 * SP * 2)             // W2Np[j/2][k] = {W2[j][k],W2[j+1][k]}
#define OFF_W1P (2 * 64 * SP * 2)          // W1P[c/2][j]  = {W1[j][c],W1[j][c+1]}
#define OFF_W1T (OFF_W1P + 2 * SP * 2)     // W1T[c][k] = W1[k][c] (row access)
#define OFF_B1 (OFF_W1T + 4 * SJ)
#define OFF_B2 (OFF_B1 + 128)
#define OFF_W3 (OFF_B2 + 128)
#define OFF_WAVE (OFF_W3 + 128)
#define WAVE_FLOATS (5 * 16 * SJ)          // h1, h2, q, ea, eb  (16 x 128 each)
#define TOTAL_LDS_BYTES ((OFF_WAVE + 4 * WAVE_FLOATS) * 4)

struct APair { v2f a0, a1; };

static __device__ inline v8f wmma4(v2f a, v2f b, v8f c) {
  // D(16x16,f32) = A(16x4,f32) x B(4x16,f32) + C
  return __builtin_amdgcn_wmma_f32_16x16x4_f32(false, a, false, b, (short)0, c,
                                               false, false);
}
static __device__ inline float xorsum16(float v) {
  return v + __shfl_xor(v, 16, 32);
}
static __device__ inline float dot4(v4f a, v4f b) {
  return a.x * b.x + a.y * b.y + a.z * b.z + a.w * b.w;
}

// D[16][128] = A(16x128 via afrag) * B(128x128 paired rows in LDS) (+bias).
// bias==nullptr -> first K-chunk uses inline-0 C (no splat movs).
template <typename AF>
static __device__ inline void gemm128(const v2f* Bp, const float* bias,
                                      float* Dout, int lr, int kh, AF afrag) {
  v8f acc[8];
  int k4start;
  if (bias) {
#pragma unroll
    for (int jt = 0; jt < 8; ++jt) {
      float bv = bias[jt * 16 + lr];
#pragma unroll
      for (int r = 0; r < 8; ++r) acc[jt][r] = bv;
    }
    k4start = 0;
  } else {
    v2f a = afrag(2 * kh);
    const v2f* bp = Bp + kh * SP + lr;
#pragma unroll
    for (int jt = 0; jt < 8; ++jt) {
      v8f z = {};
      acc[jt] = wmma4(a, bp[jt * 16], z);
    }
    k4start = 1;
  }
#pragma unroll 4
  for (int k4 = k4start; k4 < 32; ++k4) {
    const int k = k4 * 4 + 2 * kh;
    v2f a = afrag(k);
    const v2f* bp = Bp + (k4 * 2 + kh) * SP + lr;
#pragma unroll
    for (int jt = 0; jt < 8; ++jt) acc[jt] = wmma4(a, bp[jt * 16], acc[jt]);
  }
#pragma unroll
  for (int jt = 0; jt < 8; ++jt)
#pragma unroll
    for (int r = 0; r < 8; ++r)
      Dout[(r + 8 * kh) * SJ + jt * 16 + lr] = acc[jt][r];
}

// Dual-direction GEMM: one B load feeds two WMMAs. All accumulators live in
// registers until after the K-loop, so D0/D1 may overwrite buffers that the
// A-fragments read (loads precede stores in program order).
template <typename AF>
static __device__ inline void gemm128_dual(const v2f* Bp, float* D0, float* D1,
                                           int lr, int kh, AF afrag) {
  v8f acc0[8], acc1[8];
  {
    APair p = afrag(2 * kh);
    const v2f* bp = Bp + kh * SP + lr;
#pragma unroll
    for (int jt = 0; jt < 8; ++jt) {
      v2f b = bp[jt * 16];
      v8f z0 = {}, z1 = {};
      acc0[jt] = wmma4(p.a0, b, z0);
      acc1[jt] = wmma4(p.a1, b, z1);
    }
  }
#pragma unroll 2
  for (int k4 = 1; k4 < 32; ++k4) {
    const int k = k4 * 4 + 2 * kh;
    APair p = afrag(k);
    const v2f* bp = Bp + (k4 * 2 + kh) * SP + lr;
#pragma unroll
    for (int jt = 0; jt < 8; ++jt) {
      v2f b = bp[jt * 16];
      acc0[jt] = wmma4(p.a0, b, acc0[jt]);
      acc1[jt] = wmma4(p.a1, b, acc1[jt]);
    }
  }
#pragma unroll
  for (int jt = 0; jt < 8; ++jt)
#pragma unroll
    for (int r = 0; r < 8; ++r) {
      D0[(r + 8 * kh) * SJ + jt * 16 + lr] = acc0[jt][r];
      D1[(r + 8 * kh) * SJ + jt * 16 + lr] = acc1[jt][r];
    }
}

// Row-wise softmax over 128 cols for 16 rows; lane pair (lr, lr+16) covers one
// row (64 cols each half). Optionally returns sum_j w3[j]*h[j].
template <bool WSUM>
static __device__ inline float softmax_rows(float* buf, const float* w3p,
                                            int lr, int kh) {
  float* row = buf + lr * SJ + kh * 64;
  v4f* r4 = (v4f*)row;
  float mx = -3.402823466e38f;
#pragma unroll 4
  for (int i = 0; i < 16; ++i) {
    v4f v = r4[i];
    mx = fmaxf(mx, fmaxf(fmaxf(v.x, v.y), fmaxf(v.z, v.w)));
  }
  mx = fmaxf(mx, __shfl_xor(mx, 16, 32));
  float sum = 0.0f;
#pragma unroll 4
  for (int i = 0; i < 16; ++i) {
    v4f v = r4[i];
    v.x = __expf(v.x - mx); v.y = __expf(v.y - mx);
    v.z = __expf(v.z - mx); v.w = __expf(v.w - mx);
    r4[i] = v;
    sum += (v.x + v.y) + (v.z + v.w);
  }
  sum = xorsum16(sum);
  float inv = 1.0f / sum;
  float ws = 0.0f;
  const v4f* w3r = (const v4f*)(w3p + kh * 64);
#pragma unroll 4
  for (int i = 0; i < 16; ++i) {
    v4f v = r4[i];
    v.x *= inv; v.y *= inv; v.z *= inv; v.w *= inv;
    r4[i] = v;
    if (WSUM) ws += dot4(w3r[i], v);
  }
  if (WSUM) ws = xorsum16(ws);
  return ws;
}

__global__ __launch_bounds__(128, 1) void lnn_kernel(
    const float* __restrict__ x, const float* __restrict__ W1g,
    const float* __restrict__ b1g, const float* __restrict__ W2g,
    const float* __restrict__ b2g, const float* __restrict__ W3g,
    float* __restrict__ out, int Btot) {
  extern __shared__ float lds[];
  const int tid = threadIdx.x;
  const int wave = tid >> 5;
  const int lid = tid & 31;
  const int lr = lid & 15;  // sample row within wave tile
  const int kh = lid >> 4;  // half selector (K split / col split)

  // ---- cooperative weight staging (paired-row B layouts) ----
  for (int i = tid; i < 128 * 128; i += 128) {
    int j = i >> 7, k = i & 127;
    float v = W2g[i];  // W2[j][k]
    lds[OFF_W2NP + (j >> 1) * (SP * 2) + k * 2 + (j & 1)] = v;
    lds[OFF_W2TP + (k >> 1) * (SP * 2) + j * 2 + (k & 1)] = v;
  }
  if (tid < 128) {
    int k = tid;
#pragma unroll
    for (int c = 0; c < 4; ++c) {
      float v = W1g[k * 4 + c];
      lds[OFF_W1T + c * SJ + k] = v;
      lds[OFF_W1P + (c >> 1) * (SP * 2) + k * 2 + (c & 1)] = v;
    }
    lds[OFF_B1 + k] = b1g[k];
    lds[OFF_B2 + k] = b2g[k];
    lds[OFF_W3 + k] = W3g[k];
  }
  __syncthreads();

  float* h1b = &lds[OFF_WAVE + wave * WAVE_FLOATS];
  float* h2b = h1b + 16 * SJ;
  float* qb = h2b + 16 * SJ;
  float* eab = qb + 16 * SJ;   // dz2 (dir c=2), later dq (c=2)
  float* ebb = eab + 16 * SJ;  // dz2 (dir c=3), later dq (c=3)
  const float* w3p = &lds[OFF_W3];
  const int sbase = blockIdx.x * 64 + wave * 16;

  // ---- z1 = x*W1^T + b1 (single K=4 WMMA per j-tile), softmax -> h1 ----
  {
    int sidx = sbase + lr;
    if (sidx >= Btot) sidx = Btot - 1;
    v2f ax = *(const v2f*)(x + (size_t)sidx * 4 + 2 * kh);
    const v2f* bp = (const v2f*)&lds[OFF_W1P] + kh * SP + lr;
#pragma unroll
    for (int jt = 0; jt < 8; ++jt) {
      float bv = lds[OFF_B1 + jt * 16 + lr];
      v8f acc;
#pragma unroll
      for (int r = 0; r < 8; ++r) acc[r] = bv;
      acc = wmma4(ax, bp[jt * 16], acc);
#pragma unroll
      for (int r = 0; r < 8; ++r) h1b[(r + 8 * kh) * SJ + jt * 16 + lr] = acc[r];
    }
  }
  softmax_rows<false>(h1b, w3p, lr, kh);

  const float* h1row = h1b + lr * SJ;
  // ---- z2 = h1*W2^T + b2, softmax -> h2 ; s2 = w3.h2 ----
  gemm128((const v2f*)&lds[OFF_W2TP], &lds[OFF_B2], h2b, lr, kh,
          [=](int k) { return *(const v2f*)(h1row + k); });
  const float s2m = softmax_rows<true>(h2b, w3p, lr, kh);

  const float* h2row = h2b + lr * SJ;
  // ---- q = u*W2 with u = h2 .* (w3 - s2) ----
  gemm128((const v2f*)&lds[OFF_W2NP], nullptr, qb, lr, kh, [=](int k) {
    v2f h = *(const v2f*)(h2row + k);
    v2f w = *(const v2f*)(w3p + k);
    v2f a;
    a.x = h.x * (w.x - s2m);
    a.y = h.y * (w.y - s2m);
    return a;
  });

  // ---- P1: t = q.h1 ; m_r = W1col_r.h1 ; n0 = W1col_0.(h1.*q) ; g0 ----
  float tval = 0, mr0 = 0, mr1 = 0, mr2 = 0, mr3 = 0, n0 = 0;
  {
    const v4f* h4 = (const v4f*)(h1b + lr * SJ + kh * 64);
    const v4f* q4 = (const v4f*)(qb + lr * SJ + kh * 64);
    const v4f* w0 = (const v4f*)(&lds[OFF_W1T + 0 * SJ + kh * 64]);
    const v4f* w1 = (const v4f*)(&lds[OFF_W1T + 1 * SJ + kh * 64]);
    const v4f* w2 = (const v4f*)(&lds[OFF_W1T + 2 * SJ + kh * 64]);
    const v4f* w3r = (const v4f*)(&lds[OFF_W1T + 3 * SJ + kh * 64]);
#pragma unroll 4
    for (int i = 0; i < 16; ++i) {
      v4f h = h4[i], q = q4[i];
      v4f hq;
      hq.x = h.x * q.x; hq.y = h.y * q.y; hq.z = h.z * q.z; hq.w = h.w * q.w;
      tval += (hq.x + hq.y) + (hq.z + hq.w);
      mr0 += dot4(w0[i], h); mr1 += dot4(w1[i], h);
      mr2 += dot4(w2[i], h); mr3 += dot4(w3r[i], h);
      n0 += dot4(w0[i], hq);
    }
    tval = xorsum16(tval); n0 = xorsum16(n0);
    mr0 = xorsum16(mr0); mr1 = xorsum16(mr1);
    mr2 = xorsum16(mr2); mr3 = xorsum16(mr3);
  }
  const float g0 = n0 - tval * mr0;
  const float r1a = mr2, r1b = mr3;  // r1 = h1 . W1col_c == m_c
  const float* w1c2 = &lds[OFF_W1T + 2 * SJ];
  const float* w1c3 = &lds[OFF_W1T + 3 * SJ];

  // ---- dual dz2 GEMM (both directions share B = W2^T) ----
  gemm128_dual((const v2f*)&lds[OFF_W2TP], eab, ebb, lr, kh, [=](int k) {
    v2f h = *(const v2f*)(h1row + k);
    v2f wa = *(const v2f*)(w1c2 + k);
    v2f wb = *(const v2f*)(w1c3 + k);
    APair p;
    p.a0.x = h.x * (wa.x - r1a); p.a0.y = h.y * (wa.y - r1a);
    p.a1.x = h.x * (wb.x - r1b); p.a1.y = h.y * (wb.y - r1b);
    return p;
  });

  // ---- P2 (both dirs): r2 = h2.dz2 ; r3 = u.dz2 ----
  float r2a = 0, r3a = 0, r2b = 0, r3b = 0;
  {
    const v4f* h4 = (const v4f*)(h2b + lr * SJ + kh * 64);
    const v4f* ea4 = (const v4f*)(eab + lr * SJ + kh * 64);
    const v4f* eb4 = (const v4f*)(ebb + lr * SJ + kh * 64);
    const v4f* wv = (const v4f*)(w3p + kh * 64);
#pragma unroll 4
    for (int i = 0; i < 16; ++i) {
      v4f h = h4[i], ea = ea4[i], eb = eb4[i], w = wv[i];
      v4f u;
      u.x = h.x * (w.x - s2m); u.y = h.y * (w.y - s2m);
      u.z = h.z * (w.z - s2m); u.w = h.w * (w.w - s2m);
      r2a += dot4(h, ea); r3a += dot4(u, ea);
      r2b += dot4(h, eb); r3b += dot4(u, eb);
    }
    r2a = xorsum16(r2a); r3a = xorsum16(r3a);
    r2b = xorsum16(r2b); r3b = xorsum16(r3b);
  }

  // ---- dual dq GEMM (shared B = W2); outputs overwrite dz2 buffers ----
  {
    const float* earow = eab + lr * SJ;
    const float* ebrow = ebb + lr * SJ;
    gemm128_dual((const v2f*)&lds[OFF_W2NP], eab, ebb, lr, kh, [=](int k) {
      v2f h = *(const v2f*)(h2row + k);
      v2f w = *(const v2f*)(w3p + k);
      v2f ea = *(const v2f*)(earow + k);
      v2f eb = *(const v2f*)(ebrow + k);
      v2f us;
      us.x = w.x - s2m; us.y = w.y - s2m;
      APair p;
      p.a0.x = h.x * (us.x * (ea.x - r2a) - r3a);
      p.a0.y = h.y * (us.y * (ea.y - r2a) - r3a);
      p.a1.x = h.x * (us.x * (eb.x - r2b) - r3b);
      p.a1.y = h.y * (us.y * (eb.y - r2b) - r3b);
      return p;
    });
  }

  // ---- P3 (both dirs): dt, A_r ; Hcol = A_r - dt*m_r ----
  float Hcol[2][4];
  {
    float dtA = 0, dtB = 0;
    float Aa0 = 0, Aa1 = 0, Aa2 = 0, Aa3 = 0;
    float Ab0 = 0, Ab1 = 0, Ab2 = 0, Ab3 = 0;
    const v4f* h4 = (const v4f*)(h1b + lr * SJ + kh * 64);
    const v4f* q4 = (const v4f*)(qb + lr * SJ + kh * 64);
    const v4f* fa4 = (const v4f*)(eab + lr * SJ + kh * 64);
    const v4f* fb4 = (const v4f*)(ebb + lr * SJ + kh * 64);
    const v4f* wc2 = (const v4f*)(w1c2 + kh * 64);
    const v4f* wc3 = (const v4f*)(w1c3 + kh * 64);
    const v4f* w0 = (const v4f*)(&lds[OFF_W1T + 0 * SJ + kh * 64]);
    const v4f* w1 = (const v4f*)(&lds[OFF_W1T + 1 * SJ + kh * 64]);
#pragma unroll 2
    for (int i = 0; i < 16; ++i) {
      v4f h = h4[i], q = q4[i], fa = fa4[i], fb = fb4[i];
      v4f wa = wc2[i], wb = wc3[i];
      v4f dha, dhb, cma, cmb;
      dha.x = h.x * (wa.x - r1a); dha.y = h.y * (wa.y - r1a);
      dha.z = h.z * (wa.z - r1a); dha.w = h.w * (wa.w - r1a);
      dhb.x = h.x * (wb.x - r1b); dhb.y = h.y * (wb.y - r1b);
      dhb.z = h.z * (wb.z - r1b); dhb.w = h.w * (wb.w - r1b);
      cma.x = dha.x * (q.x - tval) + h.x * fa.x;
      cma.y = dha.y * (q.y - tval) + h.y * fa.y;
      cma.z = dha.z * (q.z - tval) + h.z * fa.z;
      cma.w = dha.w * (q.w - tval) + h.w * fa.w;
      cmb.x = dhb.x * (q.x - tval) + h.x * fb.x;
      cmb.y = dhb.y * (q.y - tval) + h.y * fb.y;
      cmb.z = dhb.z * (q.z - tval) + h.z * fb.z;
      cmb.w = dhb.w * (q.w - tval) + h.w * fb.w;
      Aa0 += dot4(w0[i], cma); Aa1 += dot4(w1[i], cma);
      Aa2 += dot4(wc2[i], cma); Aa3 += dot4(wc3[i], cma);
      Ab0 += dot4(w0[i], cmb); Ab1 += dot4(w1[i], cmb);
      Ab2 += dot4(wc2[i], cmb); Ab3 += dot4(wc3[i], cmb);
      dtA += dot4(fa, h) + dot4(q, dha);
      dtB += dot4(fb, h) + dot4(q, dhb);
    }
    dtA = xorsum16(dtA); dtB = xorsum16(dtB);
    Aa0 = xorsum16(Aa0); Aa1 = xorsum16(Aa1);
    Aa2 = xorsum16(Aa2); Aa3 = xorsum16(Aa3);
    Ab0 = xorsum16(Ab0); Ab1 = xorsum16(Ab1);
    Ab2 = xorsum16(Ab2); Ab3 = xorsum16(Ab3);
    Hcol[0][0] = Aa0 - dtA * mr0; Hcol[0][1] = Aa1 - dtA * mr1;
    Hcol[0][2] = Aa2 - dtA * mr2; Hcol[0][3] = Aa3 - dtA * mr3;
    Hcol[1][0] = Ab0 - dtB * mr0; Hcol[1][1] = Ab1 - dtB * mr1;
    Hcol[1][2] = Ab2 - dtB * mr2; Hcol[1][3] = Ab3 - dtB * mr3;
  }

  // ---- 2x2 solve + store (lanes 0..15; each lane = one sample) ----
  if (lid < 16) {
    int sidx = sbase + lid;
    if (sidx < Btot) {
      v2f xq = *(const v2f*)(x + (size_t)sidx * 4 + 2);
      float a = Hcol[0][2];   // H[2,2]
      float b_ = Hcol[1][2];  // H[2,3]
      float c_ = Hcol[0][3];  // H[3,2]
      float d = Hcol[1][3];   // H[3,3]
      float rhs0 = g0 - (Hcol[0][0] * xq.x + Hcol[0][1] * xq.y);
      float rhs1 = g0 - (Hcol[1][0] * xq.x + Hcol[1][1] * xq.y);
      float inv = 1.0f / (a * d - b_ * c_);
      v4f o;
      o.x = xq.x;
      o.y = xq.y;
      o.z = (d * rhs0 - b_ * rhs1) * inv;
      o.w = (-c_ * rhs0 + a * rhs1) * inv;
      *(v4f*)(out + (size_t)sidx * 4) = o;
    }
  }
}

extern "C" void kernel_launch(void* const* d_in, const int* in_sizes, int n_in,
                              void* d_out, int out_size, void* d_ws,
                              size_t ws_size, hipStream_t stream) {
  const float* x = (const float*)d_in[0];
  const float* W1 = (const float*)d_in[1];
  const float* b1 = (const float*)d_in[2];
  const float* W2 = (const float*)d_in[3];
  const float* b2 = (const float*)d_in[4];
  const float* W3 = (const float*)d_in[5];
  // b3 (d_in[6]) does not affect gradients/Hessian -> unused
  float* out = (float*)d_out;
  int B = in_sizes[0] / 4;
  int blocks = (B + 63) / 64;
  lnn_kernel<<<blocks, 128, TOTAL_LDS_BYTES, stream>>>(x, W1, b1, W2, b2, W3,
                                                       out, B);
}